// DWTInverse_66640712565475
// MI455X (gfx1250) — compile-verified
//
#include <hip/hip_runtime.h>
#include <cstdint>

// 3-level fused inverse Haar DWT (DWTInverse).
//   yl  (16,3,128,128), yh2 (16,3,3,128,128) -> LL1 256^2
//   + yh1 (16,3,3,256,256)                   -> LL2 512^2
//   + yh0 (16,3,3,512,512)                   -> out (16,3,1024,1024) f32
// Memory-bound: ~402MB fused traffic -> ~17us at 23.3 TB/s.
// Each block: one (n,c) plane, one 64x64 output tile. Intermediates in LDS.
// yh1/yh0 tiles staged via CDNA5 async global->LDS B128 loads (ASYNCcnt),
// overlapped with level-1/level-2 butterflies.

__device__ __forceinline__ void async_copy_b128(void* lds_dst, const void* gsrc) {
    // Generic pointer to __shared__ carries the LDS byte offset in its low 32 bits
    unsigned int loff = (unsigned int)(uintptr_t)lds_dst;
    asm volatile("global_load_async_to_lds_b128 %0, %1, off"
                 :: "v"(loff), "v"(gsrc)
                 : "memory");
}

__global__ __launch_bounds__(256) void haar_idwt3_fused(
    const float* __restrict__ yl,   // (48,128,128)
    const float* __restrict__ yh0,  // (48,3,512,512)
    const float* __restrict__ yh1,  // (48,3,256,256)
    const float* __restrict__ yh2,  // (48,3,128,128)
    const float* __restrict__ g0,   // (2,)
    const float* __restrict__ g1,   // (2,)
    float* __restrict__ out)        // (48,1024,1024)
{
    __shared__ __align__(16) float s_yh0[3][32][32]; // 12 KB, async target
    __shared__ __align__(16) float s_yh1[3][16][16]; //  3 KB, async target
    __shared__ __align__(16) float s_ll1[16][16];    //  1 KB
    __shared__ __align__(16) float s_ll2[32][32];    //  4 KB

    const int tid = threadIdx.x;
    const int tx = blockIdx.x;   // tile col (0..15)
    const int ty = blockIdx.y;   // tile row (0..15)
    const int nc = blockIdx.z;   // fused N*C (0..47)

    // Synthesis taps (uniform scalar loads).
    const float g00 = g0[0], g01 = g0[1];
    const float g10 = g1[0], g11 = g1[1];

    // ---- Issue async global->LDS loads: yh1 tile first, then yh0 tile ----
    // yh1 tile: 3 bands x 16x16 f32 = 192 B128 chunks -> threads 0..191, 1 each.
    const int gy1 = ty * 16, gx1 = tx * 16;
    if (tid < 192) {
        int b  = tid / 64;
        int k  = tid - b * 64;          // 0..63 chunks per band (16 rows x 4 chunks)
        int r  = k >> 2;
        int cq = (k & 3) << 2;          // col in floats (16B chunk)
        const float* src = yh1 + (((size_t)(nc * 3 + b)) << 16)
                               + (size_t)(gy1 + r) * 256 + (gx1 + cq);
        async_copy_b128(&s_yh1[b][r][cq], src);
    }
    // yh0 tile: 3 bands x 32x32 f32 = 768 B128 chunks -> 3 per thread (one/band).
    const int gy0 = ty * 32, gx0 = tx * 32;
    {
        int r  = tid >> 3;
        int cq = (tid & 7) << 2;
        size_t rowoff = (size_t)(gy0 + r) * 512 + (gx0 + cq);
        async_copy_b128(&s_yh0[0][r][cq], yh0 + (((size_t)(nc * 3 + 0)) << 18) + rowoff);
        async_copy_b128(&s_yh0[1][r][cq], yh0 + (((size_t)(nc * 3 + 1)) << 18) + rowoff);
        async_copy_b128(&s_yh0[2][r][cq], yh0 + (((size_t)(nc * 3 + 2)) << 18) + rowoff);
    }

    // ---- Level 1: yl + yh2 (8x8 sites, tiny; plain loads overlap the asyncs) ----
    if (tid < 64) {
        int i = tid >> 3, j = tid & 7;
        size_t off = (size_t)(ty * 8 + i) * 128 + (tx * 8 + j);
        float ll = yl [(((size_t)nc)            << 14) + off];
        float lh = yh2[(((size_t)(nc * 3 + 0))  << 14) + off];
        float hl = yh2[(((size_t)(nc * 3 + 1))  << 14) + off];
        float hh = yh2[(((size_t)(nc * 3 + 2))  << 14) + off];
        float lo_e = g00 * ll + g10 * lh, lo_o = g01 * ll + g11 * lh;
        float hi_e = g00 * hl + g10 * hh, hi_o = g01 * hl + g11 * hh;
        s_ll1[2 * i    ][2 * j    ] = g00 * lo_e + g10 * hi_e;
        s_ll1[2 * i    ][2 * j + 1] = g01 * lo_e + g11 * hi_e;
        s_ll1[2 * i + 1][2 * j    ] = g00 * lo_o + g10 * hi_o;
        s_ll1[2 * i + 1][2 * j + 1] = g01 * lo_o + g11 * hi_o;
    }

    // yh1 chunk was each wave's first async load; async loads complete in order,
    // so asynccnt <= 3 (the 3 yh0 loads) implies the yh1 chunk has landed.
    asm volatile("s_wait_asynccnt 0x3" ::: "memory");
    __syncthreads();   // s_ll1 + s_yh1 visible workgroup-wide

    // ---- Level 2: LL1 + yh1 -> LL2 (16x16 sites, one per thread) ----
    {
        int i = tid >> 4, j = tid & 15;
        float ll = s_ll1[i][j];
        float lh = s_yh1[0][i][j];
        float hl = s_yh1[1][i][j];
        float hh = s_yh1[2][i][j];
        float lo_e = g00 * ll + g10 * lh, lo_o = g01 * ll + g11 * lh;
        float hi_e = g00 * hl + g10 * hh, hi_o = g01 * hl + g11 * hh;
        s_ll2[2 * i    ][2 * j    ] = g00 * lo_e + g10 * hi_e;
        s_ll2[2 * i    ][2 * j + 1] = g01 * lo_e + g11 * hi_e;
        s_ll2[2 * i + 1][2 * j    ] = g00 * lo_o + g10 * hi_o;
        s_ll2[2 * i + 1][2 * j + 1] = g01 * lo_o + g11 * hi_o;
    }

    asm volatile("s_wait_asynccnt 0x0" ::: "memory");
    __syncthreads();   // s_ll2 + s_yh0 visible workgroup-wide

    // ---- Level 3: LL2 + yh0 -> output (32x32 sites, 4 horiz. sites/thread) ----
    {
        int i  = tid >> 3;            // site row 0..31
        int q  = tid & 7;             // 4-site group 0..7
        int j0 = q << 2;              // site col base

        const float4 llv = *(const float4*)&s_ll2[i][j0];
        const float4 lhv = *(const float4*)&s_yh0[0][i][j0];
        const float4 hlv = *(const float4*)&s_yh0[1][i][j0];
        const float4 hhv = *(const float4*)&s_yh0[2][i][j0];

        const float llA[4] = { llv.x, llv.y, llv.z, llv.w };
        const float lhA[4] = { lhv.x, lhv.y, lhv.z, lhv.w };
        const float hlA[4] = { hlv.x, hlv.y, hlv.z, hlv.w };
        const float hhA[4] = { hhv.x, hhv.y, hhv.z, hhv.w };

        float top[8], bot[8];
#pragma unroll
        for (int k = 0; k < 4; ++k) {
            float lo_e = g00 * llA[k] + g10 * lhA[k];
            float lo_o = g01 * llA[k] + g11 * lhA[k];
            float hi_e = g00 * hlA[k] + g10 * hhA[k];
            float hi_o = g01 * hlA[k] + g11 * hhA[k];
            top[2 * k    ] = g00 * lo_e + g10 * hi_e;
            top[2 * k + 1] = g01 * lo_e + g11 * hi_e;
            bot[2 * k    ] = g00 * lo_o + g10 * hi_o;
            bot[2 * k + 1] = g01 * lo_o + g11 * hi_o;
        }

        float* ob = out + (((size_t)nc) << 20)
                        + (size_t)(ty * 64 + 2 * i) * 1024 + (tx * 64 + (q << 3));
        *(float4*)(ob         ) = make_float4(top[0], top[1], top[2], top[3]);
        *(float4*)(ob + 4     ) = make_float4(top[4], top[5], top[6], top[7]);
        *(float4*)(ob + 1024  ) = make_float4(bot[0], bot[1], bot[2], bot[3]);
        *(float4*)(ob + 1028  ) = make_float4(bot[4], bot[5], bot[6], bot[7]);
    }
}

extern "C" void kernel_launch(void* const* d_in, const int* in_sizes, int n_in,
                              void* d_out, int out_size, void* d_ws, size_t ws_size,
                              hipStream_t stream) {
    (void)in_sizes; (void)n_in; (void)out_size; (void)d_ws; (void)ws_size;
    const float* yl  = (const float*)d_in[0];
    const float* yh0 = (const float*)d_in[1];
    const float* yh1 = (const float*)d_in[2];
    const float* yh2 = (const float*)d_in[3];
    const float* g0  = (const float*)d_in[4];
    const float* g1  = (const float*)d_in[5];
    float* out = (float*)d_out;

    dim3 grid(16, 16, 48);   // (tile_x, tile_y, N*C)
    dim3 block(256);         // 8 wave32
    haar_idwt3_fused<<<grid, block, 0, stream>>>(yl, yh0, yh1, yh2, g0, g1, out);
}